// SppPooling_17102559773029
// MI455X (gfx1250) — compile-verified
//
#include <hip/hip_runtime.h>
#include <stdint.h>

#define NGRAPH   64      // B
#define GRID_G   8       // G
#define NBINS    64      // G*G per graph
#define DIM      128     // D
#define W_CHUNKS 8       // chunks per graph
#define NWAVES   8       // waves per block (256 threads, wave32)
#define DEPTH    4       // async pipeline depth (slots per wave)

// ---------------- kernel 0: zero the output (harness poisons d_out) --------
__global__ void spp_zero(float4* out, int n4) {
    int i = blockIdx.x * blockDim.x + threadIdx.x;
    if (i < n4) out[i] = make_float4(0.f, 0.f, 0.f, 0.f);
}

// ---------------- kernel 1: exclusive prefix of batch_num_nodes ------------
__global__ void spp_prefix(const int* __restrict__ bnn, int* __restrict__ pre) {
    if (threadIdx.x == 0) {
        int s = 0;
        pre[0] = 0;
        for (int i = 0; i < NGRAPH; ++i) { s += bnn[i]; pre[i + 1] = s; }
    }
}

// ---------------- kernel 2: main scatter-add --------------------------------
__global__ __launch_bounds__(256) void spp_main(
    const float* __restrict__ feat,   // [N,128]
    const int*   __restrict__ xy,     // [N,3]
    const int*   __restrict__ pre,    // [65]
    float*       __restrict__ out)    // [64*64*128]
{
    __shared__ __align__(16) float bins[NBINS * DIM];                 // 32 KB
    __shared__ __align__(16) float stage[NWAVES * DEPTH * DIM];       // 16 KB

    const int tid  = threadIdx.x;
    const int wave = tid >> 5;
    const int lane = tid & 31;

    // zero accumulators
    for (int i = tid; i < NBINS * DIM; i += 256) bins[i] = 0.0f;
    __syncthreads();

    const int g  = blockIdx.x / W_CHUNKS;
    const int ck = blockIdx.x % W_CHUNKS;
    const int s0 = pre[g];
    const int s1 = pre[g + 1];
    const int len = s1 - s0;
    const int beg = s0 + (int)(((long long)len * ck) / W_CHUNKS);
    const int end = s0 + (int)(((long long)len * (ck + 1)) / W_CHUNKS);

    float* mystage = &stage[wave * DEPTH * DIM];
    const uint32_t lds_base = (uint32_t)(uintptr_t)mystage;  // low 32 bits == LDS offset

    const int node0 = beg + wave;

    if (node0 < end) {
        // ---- prime the async pipeline: DEPTH outstanding loads, clamped so
        //      the in-flight count stays constant (immediate-operand waits). ----
        #pragma unroll
        for (int p = 0; p < DEPTH; ++p) {
            int pn = node0 + p * NWAVES;
            if (pn >= end) pn = node0;  // dummy (never consumed as node pn)
            unsigned long long ga = (unsigned long long)(uintptr_t)(feat + (size_t)pn * DIM)
                                  + (unsigned long long)(lane << 4);
            uint32_t la = lds_base + (uint32_t)(p * (DIM * 4)) + (uint32_t)(lane << 4);
            asm volatile("global_load_async_to_lds_b128 %0, %1, off"
                         :: "v"(la), "v"(ga) : "memory");
        }

        int slot = 0;
        for (int n = node0; n < end; n += NWAVES) {
            // oldest async load (this node's row) has landed in LDS
            asm volatile("s_wait_asynccnt 0x3" ::: "memory");

            const float4 v = *(const float4*)(mystage + slot * DIM + (lane << 2));

            const int r  = xy[3 * n + 0];
            const int cc = xy[3 * n + 1];
            const int dv = xy[3 * n + 2];
            const float w = __builtin_amdgcn_rcpf((float)dv);

            const float a0 = v.x * w, a1 = v.y * w, a2 = v.z * w, a3 = v.w * w;

            // WAR hazard: DS read of this slot must retire before the async
            // engine overwrites it (different pipes, not mutually ordered).
            asm volatile("s_wait_dscnt 0x0" ::: "memory");

            int pn = n + DEPTH * NWAVES;
            if (pn >= end) pn = node0;  // keep in-flight count constant
            unsigned long long ga = (unsigned long long)(uintptr_t)(feat + (size_t)pn * DIM)
                                  + (unsigned long long)(lane << 4);
            uint32_t la = lds_base + (uint32_t)(slot * (DIM * 4)) + (uint32_t)(lane << 4);
            asm volatile("global_load_async_to_lds_b128 %0, %1, off"
                         :: "v"(la), "v"(ga) : "memory");

            float* b = &bins[(((r << 3) + cc) * DIM) + (lane << 2)];
            atomicAdd(b + 0, a0);   // ds_add_f32
            atomicAdd(b + 1, a1);
            atomicAdd(b + 2, a2);
            atomicAdd(b + 3, a3);

            slot = (slot + 1) & (DEPTH - 1);
        }
        // drain dangling (clamped) prefetches
        asm volatile("s_wait_asynccnt 0x0" ::: "memory");
    }

    __syncthreads();

    // ---- flush 64 bins x 128 cols to global with hardware f32 atomics ----
    float* obase = out + (size_t)g * (NBINS * DIM);
    for (int i = tid; i < NBINS * DIM; i += 256) {
        unsafeAtomicAdd(obase + i, bins[i]);   // global_atomic_add_f32
    }
}

// ---------------- host-side launcher ----------------------------------------
extern "C" void kernel_launch(void* const* d_in, const int* in_sizes, int n_in,
                              void* d_out, int out_size, void* d_ws, size_t ws_size,
                              hipStream_t stream) {
    const float* feat = (const float*)d_in[0];   // [N*128] fp32
    const int*   xy   = (const int*)d_in[1];     // [N*3]   int32
    const int*   bnn  = (const int*)d_in[2];     // [64]    int32
    float*       out  = (float*)d_out;           // [64*8*8*128] fp32
    int*         pre  = (int*)d_ws;              // [65] scratch

    const int n4 = out_size / 4;                 // 131072 float4
    spp_zero<<<(n4 + 255) / 256, 256, 0, stream>>>((float4*)out, n4);
    spp_prefix<<<1, 64, 0, stream>>>(bnn, pre);
    spp_main<<<NGRAPH * W_CHUNKS, 256, 0, stream>>>(feat, xy, pre, out);
}